// Attention_86105504350880
// MI455X (gfx1250) — compile-verified
//
#include <hip/hip_runtime.h>

#define DIMX 4096
#define NH   32
#define DH   128
#define PAST 4096
#define NPOS (PAST + 1)

typedef __attribute__((ext_vector_type(2))) float v2f;
typedef __attribute__((ext_vector_type(8))) float v8f;

// ---------------------------------------------------------------------------
// WMMA-based 16-row GEMV tile: out[tile*16 .. +15] = W[rows,:] . x
// A (16x4 f32): lane L(0..15) holds W[row=L, k0..k1]; lane L+16 holds k2..k3
// B (4x16 f32): broadcast of x chunk -> every column of D = the 16 dot products
// K (4096) is split across the block's 4 waves; partials combined in LDS.
// ---------------------------------------------------------------------------
__device__ __forceinline__ void gemv16_tile(const float* __restrict__ W,
                                            const float* __restrict__ xs,   // LDS copy of x
                                            float* __restrict__ partial,    // LDS [64]
                                            float* __restrict__ out,
                                            int tile)
{
    const int tid   = threadIdx.x;
    const int lane  = tid & 31;
    const int wave  = tid >> 5;            // 0..3 : K split
    const int rowIT = lane & 15;
    const int kHalf = (lane >> 4) << 1;    // 0 or 2
    const int row   = tile * 16 + rowIT;
    const int kbeg  = wave * (DIMX / 4);   // 1024 per wave

    const float* wp = W + (size_t)row * DIMX + kbeg + kHalf;
    const float* xp = xs + kbeg + kHalf;

    v8f c = {};
#pragma unroll 4
    for (int i = 0; i < DIMX / 16; ++i) {      // 256 iters, 4 K per iter
        __builtin_prefetch(wp + 128, 0, 3);    // global_prefetch_b8, ~512B ahead
        v2f a, b;
        a[0] = wp[0]; a[1] = wp[1];
        b[0] = xp[0]; b[1] = xp[1];
        c = __builtin_amdgcn_wmma_f32_16x16x4_f32(false, a, false, b,
                                                  (short)0, c, false, false);
        wp += 4; xp += 4;
    }
    // D layout: VGPR r -> M=r (lanes 0-15), M=8+r (lanes 16-31); column N = lane.
    // Column 0 (lane 0 / lane 16) carries all 16 results.
    if (lane == 0) {
#pragma unroll
        for (int j = 0; j < 8; ++j) partial[wave * 16 + j] = c[j];
    }
    if (lane == 16) {
#pragma unroll
        for (int j = 0; j < 8; ++j) partial[wave * 16 + 8 + j] = c[j];
    }
    __syncthreads();
    if (tid < 16) {
        out[tile * 16 + tid] =
            partial[tid] + partial[16 + tid] + partial[32 + tid] + partial[48 + tid];
    }
}

// ---------------------------------------------------------------------------
// Fused QKV projection: blockIdx selects matrix (0..2) and 16-row tile (0..255)
// ---------------------------------------------------------------------------
__global__ void __launch_bounds__(128)
qkv_kernel(const float* __restrict__ wq, const float* __restrict__ wk,
           const float* __restrict__ wv, const float* __restrict__ x,
           float* __restrict__ q, float* __restrict__ k, float* __restrict__ v)
{
    __shared__ float xs[DIMX];
    __shared__ float partial[64];
    const int tid = threadIdx.x;
    const float4* x4  = (const float4*)x;
    float4*       xs4 = (float4*)xs;
#pragma unroll
    for (int i = tid; i < DIMX / 4; i += 128) xs4[i] = x4[i];
    __syncthreads();

    const int mat  = blockIdx.x >> 8;
    const int tile = blockIdx.x & 255;
    const float* W   = (mat == 0) ? wq : (mat == 1) ? wk : wv;
    float*       out = (mat == 0) ? q  : (mat == 1) ? k  : v;
    gemv16_tile(W, xs, partial, out, tile);
}

__global__ void __launch_bounds__(128)
wo_kernel(const float* __restrict__ W, const float* __restrict__ xin,
          float* __restrict__ out)
{
    __shared__ float xs[DIMX];
    __shared__ float partial[64];
    const int tid = threadIdx.x;
    const float4* x4  = (const float4*)xin;
    float4*       xs4 = (float4*)xs;
#pragma unroll
    for (int i = tid; i < DIMX / 4; i += 128) xs4[i] = x4[i];
    __syncthreads();
    gemv16_tile(W, xs, partial, out, blockIdx.x);
}

// ---------------------------------------------------------------------------
// RoPE (in-place on q and k): 2048 pairs each, cos/sin shared across heads
// ---------------------------------------------------------------------------
__global__ void __launch_bounds__(256)
rope_kernel(float* __restrict__ q, float* __restrict__ k,
            const float* __restrict__ cosr, const float* __restrict__ sinr)
{
    const int t  = blockIdx.x * blockDim.x + threadIdx.x;   // 0..4095
    float* buf   = (t < 2048) ? q : k;
    const int pi = t & 2047;                                // pair index
    const int j  = pi & 63;                                 // pair within head
    const int base = ((pi >> 6) << 7) + 2 * j;              // h*128 + 2*j
    const float c = cosr[j], s = sinr[j];
    const float te = buf[base], to = buf[base + 1];
    buf[base]     = te * c - to * s;
    buf[base + 1] = te * s + to * c;
}

// ---------------------------------------------------------------------------
// Attention: one block per head. scores -> softmax -> weighted V
// cache layout: [H, DH, PAST], contiguous in position p.
// ---------------------------------------------------------------------------
__global__ void __launch_bounds__(512)
attn_kernel(const int* __restrict__ cacheK, const int* __restrict__ cacheV,
            const float* __restrict__ scbK, const float* __restrict__ scbV,
            const float* __restrict__ q, const float* __restrict__ k,
            const float* __restrict__ v, float* __restrict__ attn_out)
{
    const int h   = blockIdx.x;
    const int tid = threadIdx.x;                   // 512 threads = 16 waves
    const float scale   = 0.08838834764831845f;    // 1/sqrt(128)
    const float invQ127 = 1.0f / 127.0f;

    __shared__ float qs[DH];           // q pre-scaled by scb_k/127
    __shared__ float probs[NPOS + 7];  // scores -> probabilities
    __shared__ float red[512];

    if (tid < DH) qs[tid] = q[h * DH + tid] * scbK[h * DH + tid] * invQ127;
    __syncthreads();

    // ---- scores over past positions: each thread owns 8 consecutive p ----
    const int* Kh = cacheK + (size_t)h * DH * PAST;
    const int  p0 = tid * 8;
    float acc[8];
#pragma unroll
    for (int i = 0; i < 8; ++i) acc[i] = 0.0f;
    for (int d = 0; d < DH; ++d) {
        const int4* kp = (const int4*)(Kh + (size_t)d * PAST + p0);
        const int4 a = kp[0];
        const int4 b = kp[1];
        const float qd = qs[d];
        acc[0] += qd * (float)a.x;  acc[1] += qd * (float)a.y;
        acc[2] += qd * (float)a.z;  acc[3] += qd * (float)a.w;
        acc[4] += qd * (float)b.x;  acc[5] += qd * (float)b.y;
        acc[6] += qd * (float)b.z;  acc[7] += qd * (float)b.w;
    }
#pragma unroll
    for (int i = 0; i < 8; ++i) probs[p0 + i] = acc[i] * scale;
    if (tid == 0) {                 // score for the new (roped) token
        float s = 0.0f;
        for (int d = 0; d < DH; ++d) s += q[h * DH + d] * k[h * DH + d];
        probs[PAST] = s * scale;
    }
    __syncthreads();

    // ---- softmax over NPOS ----
    float m = -3.4e38f;
    for (int i = tid; i < NPOS; i += 512) m = fmaxf(m, probs[i]);
    red[tid] = m; __syncthreads();
    for (int s2 = 256; s2 > 0; s2 >>= 1) {
        if (tid < s2) red[tid] = fmaxf(red[tid], red[tid + s2]);
        __syncthreads();
    }
    const float gmax = red[0];
    __syncthreads();
    float lsum = 0.0f;
    for (int i = tid; i < NPOS; i += 512) {
        const float e = __expf(probs[i] - gmax);
        probs[i] = e;
        lsum += e;
    }
    red[tid] = lsum; __syncthreads();
    for (int s2 = 256; s2 > 0; s2 >>= 1) {
        if (tid < s2) red[tid] += red[tid + s2];
        __syncthreads();
    }
    const float inv = 1.0f / red[0];
    __syncthreads();
    for (int i = tid; i < NPOS; i += 512) probs[i] *= inv;
    __syncthreads();

    // ---- out[d] = sum_p probs[p] * V[h,d,p]*scb/127 + probs[PAST]*v_new[d] ----
    const int wave = tid >> 5, lane = tid & 31;
    const int* Vh = cacheV + (size_t)h * DH * PAST;
#pragma unroll
    for (int dd = 0; dd < 8; ++dd) {
        const int d = wave * 8 + dd;
        const int* Vrow = Vh + (size_t)d * PAST;
        float a = 0.0f;
        for (int it = 0; it < PAST / 128; ++it) {
            const int p = it * 128 + lane * 4;
            const int4 vv = *(const int4*)(Vrow + p);
            const float4 pr = *(const float4*)&probs[p];
            a += pr.x * (float)vv.x + pr.y * (float)vv.y +
                 pr.z * (float)vv.z + pr.w * (float)vv.w;
        }
#pragma unroll
        for (int off = 16; off > 0; off >>= 1) a += __shfl_down(a, off, 32);
        if (lane == 0) {
            attn_out[h * DH + d] = a * scbV[h * DH + d] * invQ127
                                 + probs[PAST] * v[h * DH + d];
        }
    }
}

// ---------------------------------------------------------------------------
extern "C" void kernel_launch(void* const* d_in, const int* in_sizes, int n_in,
                              void* d_out, int out_size, void* d_ws, size_t ws_size,
                              hipStream_t stream) {
    const float* x      = (const float*)d_in[0];
    const float* wq     = (const float*)d_in[1];
    const float* wk     = (const float*)d_in[2];
    const float* wv     = (const float*)d_in[3];
    const float* wo     = (const float*)d_in[4];
    const float* fcos   = (const float*)d_in[5];
    const float* fsin   = (const float*)d_in[6];
    const float* scbK   = (const float*)d_in[7];
    const float* scbV   = (const float*)d_in[8];
    const int*   cacheK = (const int*)d_in[9];
    const int*   cacheV = (const int*)d_in[10];
    float* out = (float*)d_out;

    float* F  = (float*)d_ws;
    float* q  = F;              // 4096
    float* k  = F + DIMX;       // 4096
    float* v  = F + 2 * DIMX;   // 4096
    float* ao = F + 3 * DIMX;   // 4096 attention output (h-major)

    // 1) fused QKV GEMV (WMMA f32 16x16x4)
    qkv_kernel<<<dim3(3 * (DIMX / 16)), dim3(128), 0, stream>>>(wq, wk, wv, x, q, k, v);
    // 2) RoPE on q and k
    rope_kernel<<<dim3(16), dim3(256), 0, stream>>>(q, k, fcos, fsin);
    // 3) attention per head
    attn_kernel<<<dim3(NH), dim3(512), 0, stream>>>(cacheK, cacheV, scbK, scbV,
                                                    q, k, v, ao);
    // 4) output projection into d_out (WMMA f32 16x16x4)
    wo_kernel<<<dim3(DIMX / 16), dim3(128), 0, stream>>>(wo, ao, out);
}